// MaskAdaIN_43542378447212
// MI455X (gfx1250) — compile-verified
//
#include <hip/hip_runtime.h>
#include <hip/hip_bf16.h>

typedef __attribute__((ext_vector_type(2))) float v2f;
typedef __attribute__((ext_vector_type(8))) float v8f;

#define EPS_ 1e-8f
#define B_ 4
#define C_ 128
#define H_ 256
#define W_ 256
#define MH_ 128
#define MW_ 128
#define HW_ (H_ * W_)            // 65536
#define PLANES_ (B_ * C_)        // 512
#define F4_PER_PLANE_ (HW_ / 4)  // 16384
#define TWO_C_ (2 * C_)          // 256

// workspace layout (floats)
#define MASK_OFF_ 0                      // 2*B_ floats: per-batch {sum(m_up), sum(m_up^2)}
#define SUMS_OFF_ 8                      // PLANES_*12 floats
#define PARAMS_OFF_ (8 + PLANES_ * 12)   // PLANES_*4 floats {scale_in, bias_in, scale_out, bias_out}

__device__ __forceinline__ float wave_red(float v) {
    v += __shfl_down(v, 16, 32);
    v += __shfl_down(v, 8, 32);
    v += __shfl_down(v, 4, 32);
    v += __shfl_down(v, 2, 32);
    v += __shfl_down(v, 1, 32);
    return v;
}

// ---------------------------------------------------------------------------
// Kernel 1: per-(b,c) masked-statistics sums (blocks 0..511) and per-batch
// upsampled-mask sums (blocks 512..515). Memory-bound streaming pass.
// Per plane we accumulate for w in {m, 1-m}, x in {x1, x2}:
//   S_a = sum(x*w), S_b = sum(x*w^3), S_c = sum(x^2*w^4)
// which are sufficient for mean and var = (S_c - 2*mean*S_b + mean^2*Sum(w^2))/n
// ---------------------------------------------------------------------------
__global__ __launch_bounds__(256) void stats_kernel(const float* __restrict__ x1,
                                                    const float* __restrict__ x2,
                                                    const float* __restrict__ mask,
                                                    float* __restrict__ ws) {
    __shared__ float part[8][12];
    const int bid = blockIdx.x;
    const int tid = threadIdx.x;
    const int wv = tid >> 5;
    const int ln = tid & 31;

    if (bid >= PLANES_) {
        // per-batch mask sums over the RAW 128x128 mask; 2x2 nearest upsample
        // means sum(m_up) = 4*sum(m_raw), sum(m_up^2) = 4*sum(m_raw^2).
        const int b = bid - PLANES_;
        const float* mp = mask + (size_t)b * (MH_ * MW_);
        float sm = 0.f, sm2 = 0.f;
        for (int i = tid; i < MH_ * MW_; i += 256) {
            const float m = mp[i];
            sm += m;
            sm2 = fmaf(m, m, sm2);
        }
        sm = wave_red(sm);
        sm2 = wave_red(sm2);
        if (ln == 0) { part[wv][0] = sm; part[wv][1] = sm2; }
        __syncthreads();
        if (tid == 0) {
            float a = 0.f, c = 0.f;
            for (int w = 0; w < 8; ++w) { a += part[w][0]; c += part[w][1]; }
            ws[MASK_OFF_ + 2 * b + 0] = 4.0f * a;
            ws[MASK_OFF_ + 2 * b + 1] = 4.0f * c;
        }
        return;
    }

    const int b = bid >> 7;  // bid = b*C_ + c
    const float4* x1p = (const float4*)(x1 + (size_t)bid * HW_);
    const float4* x2p = (const float4*)(x2 + (size_t)bid * HW_);
    const float* mp = mask + (size_t)b * (MH_ * MW_);

    float s[12];
#pragma unroll
    for (int j = 0; j < 12; ++j) s[j] = 0.f;

    for (int i = tid; i < F4_PER_PLANE_; i += 256) {
        const int h = i >> 6;        // 64 float4 per row (W_=256)
        const int w4 = i & 63;
        const float2 mf = *(const float2*)(mp + (h >> 1) * MW_ + (w4 << 1));
        const float4 xa4 = x1p[i];
        const float4 xb4 = x2p[i];
        const float mA[2] = {mf.x, mf.y};
#pragma unroll
        for (int j = 0; j < 2; ++j) {
            const float m = mA[j];
            const float om = 1.f - m;
            const float m2 = m * m, m3 = m2 * m, m4 = m2 * m2;
            const float o2 = om * om, o3 = o2 * om, o4 = o2 * o2;
#pragma unroll
            for (int e = 0; e < 2; ++e) {
                const float xa = (&xa4.x)[j * 2 + e];
                const float xb = (&xb4.x)[j * 2 + e];
                const float xa2 = xa * xa;
                const float xb2 = xb * xb;
                s[0] = fmaf(xa, m, s[0]);   s[1] = fmaf(xa, m3, s[1]);  s[2] = fmaf(xa2, m4, s[2]);
                s[3] = fmaf(xa, om, s[3]);  s[4] = fmaf(xa, o3, s[4]);  s[5] = fmaf(xa2, o4, s[5]);
                s[6] = fmaf(xb, m, s[6]);   s[7] = fmaf(xb, m3, s[7]);  s[8] = fmaf(xb2, m4, s[8]);
                s[9] = fmaf(xb, om, s[9]);  s[10] = fmaf(xb, o3, s[10]); s[11] = fmaf(xb2, o4, s[11]);
            }
        }
    }

#pragma unroll
    for (int j = 0; j < 12; ++j) {
        const float r = wave_red(s[j]);
        if (ln == 0) part[wv][j] = r;
    }
    __syncthreads();
    if (tid < 12) {
        float t = 0.f;
        for (int w = 0; w < 8; ++w) t += part[w][tid];
        ws[SUMS_OFF_ + bid * 12 + tid] = t;
    }
}

// ---------------------------------------------------------------------------
// Kernel 2: one 1024-thread workgroup.
//  Phase A: finalize 8 stats per (b,c) into LDS (v-vectors [B,2C] per gemm).
//  Phase B: 32 waves; wave (g,t) computes a 16x16 D tile of gemm g via
//           V_WMMA_F32_16X16X4_F32 (M=B padded to 16, N=16*t.., K=2C=256).
//  Phase C: fold stats + modulation into per-(b,c) {scale,bias} pairs.
// ---------------------------------------------------------------------------
__global__ __launch_bounds__(1024) void finalize_modulate_kernel(
    const float* __restrict__ wInMean, const float* __restrict__ wInVar,
    const float* __restrict__ wOutMean, const float* __restrict__ wOutVar,
    float* __restrict__ ws) {
    __shared__ float smem[8192];
    float* vArrs = smem;              // 4 gemms x [B_ x 2C] = 4*1024
    float* meanIn = smem + 4096;      // 512
    float* istdIn = smem + 4608;      // 512
    float* meanOut = smem + 5120;     // 512
    float* istdOut = smem + 5632;     // 512
    float* ada = smem + 6144;         // 4 x 512

    const int tid = threadIdx.x;

    // ---- Phase A: finalize stats ----
    if (tid < PLANES_) {
        const float* s = ws + SUMS_OFF_ + tid * 12;
        const int b = tid >> 7;
        const int c = tid & 127;
        const float Sm = ws[MASK_OFF_ + 2 * b + 0];
        const float Sm2 = ws[MASK_OFF_ + 2 * b + 1];
        const float nIn = Sm + EPS_;
        const float nOut = (float)HW_ - Sm + EPS_;
        const float Som2 = (float)HW_ - 2.f * Sm + Sm2;

        const float m1i = s[0] / nIn;
        const float v1i = fmaxf((s[2] - 2.f * m1i * s[1] + m1i * m1i * Sm2) / nIn, 0.f);
        const float m1o = s[3] / nOut;
        const float v1o = fmaxf((s[5] - 2.f * m1o * s[4] + m1o * m1o * Som2) / nOut, 0.f);
        const float m2i = s[6] / nIn;
        const float v2i = fmaxf((s[8] - 2.f * m2i * s[7] + m2i * m2i * Sm2) / nIn, 0.f);
        const float m2o = s[9] / nOut;
        const float v2o = fmaxf((s[11] - 2.f * m2o * s[10] + m2o * m2o * Som2) / nOut, 0.f);

        // v = concat([x1_stat, x2_stat]) along channel, per gemm g:
        // g0: in_mean, g1: in_var, g2: out_mean, g3: out_var
        vArrs[0 * 1024 + b * TWO_C_ + c] = m1i;  vArrs[0 * 1024 + b * TWO_C_ + C_ + c] = m2i;
        vArrs[1 * 1024 + b * TWO_C_ + c] = v1i;  vArrs[1 * 1024 + b * TWO_C_ + C_ + c] = v2i;
        vArrs[2 * 1024 + b * TWO_C_ + c] = m1o;  vArrs[2 * 1024 + b * TWO_C_ + C_ + c] = m2o;
        vArrs[3 * 1024 + b * TWO_C_ + c] = v1o;  vArrs[3 * 1024 + b * TWO_C_ + C_ + c] = v2o;
        meanIn[tid] = m1i;  istdIn[tid] = rsqrtf(v1i + EPS_);
        meanOut[tid] = m1o; istdOut[tid] = rsqrtf(v1o + EPS_);
    }
    __syncthreads();

    // ---- Phase B: 4 GEMMs [16(pad)x256]x[256x128] via WMMA f32 16x16x4 ----
    const int wv = tid >> 5;
    const int ln = tid & 31;
    const int g = wv >> 3;       // which gemm (0..3)
    const int t = wv & 7;        // which 16-wide N tile (0..7)
    const int n = ln & 15;       // A row (M) for lanes; D column (N)
    const int hi = ln >> 4;      // lane-half selects K pair {2*hi, 2*hi+1}
    const float* Wg = (g == 0) ? wInMean : (g == 1) ? wInVar : (g == 2) ? wOutMean : wOutVar;
    const float* vA = vArrs + g * 1024;

    // rows >= B_ are padding: clamp address (always in-bounds) and zero by mask
    const int rowA = (n < B_) ? n : 0;
    const float rmask = (n < B_) ? 1.f : 0.f;

#if __has_builtin(__builtin_amdgcn_wmma_f32_16x16x4_f32)
    v8f acc = {0.f, 0.f, 0.f, 0.f, 0.f, 0.f, 0.f, 0.f};
    for (int kb = 0; kb < TWO_C_; kb += 4) {
        const int k0 = kb + 2 * hi;
        v2f av, bv;
        // A (16x4 f32): lane = M, VGPR0/1 = K{0,1} lanes 0-15, K{2,3} lanes 16-31
        av.x = vA[rowA * TWO_C_ + k0] * rmask;
        av.y = vA[rowA * TWO_C_ + k0 + 1] * rmask;
        // B (4x16 f32): lane = N, mirrored K striping; B[k][c] = W[c][k]
        bv.x = Wg[(t * 16 + n) * TWO_C_ + k0];
        bv.y = Wg[(t * 16 + n) * TWO_C_ + k0 + 1];
        acc = __builtin_amdgcn_wmma_f32_16x16x4_f32(
            /*neg_a=*/false, av, /*neg_b=*/false, bv,
            /*c_mod=*/(short)0, acc, /*reuse_a=*/false, /*reuse_b=*/false);
    }
    // D: VGPR r, lanes 0-15 hold M=r, lanes 16-31 hold M=r+8; only M<B_ valid
    if (hi == 0) {
#pragma unroll
        for (int r = 0; r < B_; ++r)
            ada[g * 512 + r * C_ + t * 16 + n] = acc[r];
    }
#else
    // scalar fallback: 64 outputs per wave, 2 per lane
    for (int rr = 0; rr < 2; ++rr) {
        const int o = ln + 32 * rr;
        const int ob = o >> 4;
        const int oc = t * 16 + (o & 15);
        float d = 0.f;
        for (int k = 0; k < TWO_C_; ++k)
            d = fmaf(vA[ob * TWO_C_ + k], Wg[oc * TWO_C_ + k], d);
        ada[g * 512 + ob * C_ + oc] = d;
    }
    (void)rowA; (void)rmask; (void)hi; (void)n;
#endif
    __syncthreads();

    // ---- Phase C: fold into scale/bias so the apply pass is 2 fma chains ----
    if (tid < PLANES_) {
        const float aim = ada[0 * 512 + tid];
        const float aiv = ada[1 * 512 + tid];
        const float aom = ada[2 * 512 + tid];
        const float aov = ada[3 * 512 + tid];
        const float si = istdIn[tid] * aiv;
        const float bi = aim - meanIn[tid] * si;
        const float so = istdOut[tid] * aov;
        const float bo = aom - meanOut[tid] * so;
        float4 pv;
        pv.x = si; pv.y = bi; pv.z = so; pv.w = bo;
        ((float4*)(ws + PARAMS_OFF_))[tid] = pv;
    }
}

// ---------------------------------------------------------------------------
// Kernel 3: elementwise apply — read x1 + mask, write output. Memory-bound.
// out = (x*m*s_in + b_in)*m + (x*om*s_out + b_out)*om
// ---------------------------------------------------------------------------
__global__ __launch_bounds__(256) void apply_kernel(const float* __restrict__ x1,
                                                    const float* __restrict__ mask,
                                                    const float* __restrict__ ws,
                                                    float* __restrict__ out) {
    const int bid = blockIdx.x;
    const int tid = threadIdx.x;
    const int p = bid >> 2;   // plane (b*C + c)
    const int q = bid & 3;    // quarter of the plane
    const int b = p >> 7;
    const float4 pv = ((const float4*)(ws + PARAMS_OFF_))[p];
    const float4* xp = (const float4*)(x1 + (size_t)p * HW_);
    float4* op = (float4*)(out + (size_t)p * HW_);
    const float* mp = mask + (size_t)b * (MH_ * MW_);

#pragma unroll 4
    for (int it = 0; it < 16; ++it) {
        const int i = q * 4096 + it * 256 + tid;
        const int h = i >> 6;
        const int w4 = i & 63;
        const float2 mf = *(const float2*)(mp + (h >> 1) * MW_ + (w4 << 1));
        const float4 x = xp[i];
        float4 r;
        const float mA[2] = {mf.x, mf.y};
#pragma unroll
        for (int j = 0; j < 2; ++j) {
            const float m = mA[j];
            const float om = 1.f - m;
#pragma unroll
            for (int e = 0; e < 2; ++e) {
                const float xv = (&x.x)[j * 2 + e];
                const float ri = fmaf(xv * m, pv.x, pv.y);
                const float ro = fmaf(xv * om, pv.z, pv.w);
                (&r.x)[j * 2 + e] = fmaf(ri, m, ro * om);
            }
        }
        op[i] = r;
    }
}

extern "C" void kernel_launch(void* const* d_in, const int* in_sizes, int n_in,
                              void* d_out, int out_size, void* d_ws, size_t ws_size,
                              hipStream_t stream) {
    (void)in_sizes; (void)n_in; (void)out_size; (void)ws_size;
    const float* x1 = (const float*)d_in[0];
    const float* x2 = (const float*)d_in[1];
    const float* mask = (const float*)d_in[2];
    const float* wInMean = (const float*)d_in[3];
    const float* wInVar = (const float*)d_in[4];
    const float* wOutMean = (const float*)d_in[5];
    const float* wOutVar = (const float*)d_in[6];
    float* ws = (float*)d_ws;
    float* out = (float*)d_out;

    stats_kernel<<<PLANES_ + B_, 256, 0, stream>>>(x1, x2, mask, ws);
    finalize_modulate_kernel<<<1, 1024, 0, stream>>>(wInMean, wInVar, wOutMean, wOutVar, ws);
    apply_kernel<<<PLANES_ * 4, 256, 0, stream>>>(x1, mask, ws, out);
}